// SVDPlusPlusNet_76046690943220
// MI455X (gfx1250) — compile-verified
//
#include <hip/hip_runtime.h>

#ifndef __has_builtin
#define __has_builtin(x) 0
#endif

#define SVDPP_D 128
#define SVDPP_L 200
#define WAVES_PER_BLOCK 4
#define BLOCK_THREADS (WAVES_PER_BLOCK * 32)
#define ROWS_PER_TDM 8

// TDM gather path: CDNA5 Tensor Data Mover in gather mode (8 x 32-bit row
// indices per descriptor) — purpose-built for embedding-row gathers.
#if defined(__AMDGCN__) && __has_builtin(__builtin_amdgcn_tensor_load_to_lds) && \
    __has_builtin(__builtin_amdgcn_s_wait_tensorcnt)
#define USE_TDM 1
#else
#define USE_TDM 0
#endif

typedef unsigned int u32x4 __attribute__((ext_vector_type(4)));
typedef int i32x4 __attribute__((ext_vector_type(4)));
typedef int i32x8 __attribute__((ext_vector_type(8)));

__global__ __launch_bounds__(BLOCK_THREADS) void svdpp_score_kernel(
    const int* __restrict__ user_ids,
    const int* __restrict__ item_ids,
    const int* __restrict__ hist_items,
    const int* __restrict__ hist_len,
    const float* __restrict__ user_emb,
    const float* __restrict__ item_emb,
    const float* __restrict__ implicit_emb,
    const float* __restrict__ user_bias,
    const float* __restrict__ item_bias,
    const float* __restrict__ global_bias,
    float* __restrict__ out,
    int batch, int n_items)
{
#if USE_TDM
  // Per-wave double-buffered staging: 2 buffers x 8 rows x 128 floats = 8 KB/wave.
  __shared__ float stage[WAVES_PER_BLOCK * 2 * ROWS_PER_TDM * SVDPP_D];
#endif
  const int wave = threadIdx.x >> 5;   // wave32
  const int lane = threadIdx.x & 31;
  const int b = blockIdx.x * WAVES_PER_BLOCK + wave;
  if (b >= batch) return;

  const int len = hist_len[b];
  const int uid = user_ids[b];
  const int iid = item_ids[b];

  // Each lane owns 4 consecutive dims: 16B-aligned b128 loads.
  const float4 p = *reinterpret_cast<const float4*>(
      user_emb + (size_t)uid * SVDPP_D + lane * 4);
  const float4 q = *reinterpret_cast<const float4*>(
      item_emb + (size_t)iid * SVDPP_D + lane * 4);

  const int* __restrict__ hrow = hist_items + (size_t)b * SVDPP_L;

  float ax = 0.f, ay = 0.f, az = 0.f, aw = 0.f;

#if USE_TDM
  const unsigned wave_lds_base =
      (unsigned)(wave * 2 * ROWS_PER_TDM * SVDPP_D) * 4u;  // byte offset in LDS
  const unsigned long long gaddr = (unsigned long long)(uintptr_t)implicit_emb;

  auto tdm_issue = [&](int j0, int bsel) {
    const int nv = min(ROWS_PER_TDM, len - j0);
    int id[8];
#pragma unroll
    for (int k = 0; k < 8; ++k)
      id[k] = (k < nv) ? hrow[j0 + k] : 0;
    // D# group0: count=1 | gather_index_size=32b (bit30) | gather_mode (bit31);
    // lds_addr[63:32]; global_addr[120:64]; type=2 in [127:126].
    u32x4 g0 = {
        1u | (1u << 30) | (1u << 31),
        wave_lds_base + (unsigned)bsel * (ROWS_PER_TDM * SVDPP_D * 4u),
        (unsigned)gaddr,
        ((unsigned)(gaddr >> 32) & 0x01FFFFFFu) | 0x80000000u
    };
    // D# group1: wg_mask=0, data_size=2 (4B); tensor_dim0=128;
    // tensor_dim1=n_items (OOB bound); tile_dim0=128; tile_dim1=nv (#indices);
    // tensor_dim0_stride=128 elements. tile_dim2/tensor_dim1_stride ignored
    // in gather mode.
    i32x8 g1 = {
        (int)0x00020000u,
        (int)(SVDPP_D << 16),
        (int)(((unsigned)n_items & 0xFFFFu) << 16),
        (int)(((unsigned)SVDPP_D << 16) | ((unsigned)n_items >> 16)),
        nv,
        SVDPP_D,
        0, 0
    };
    i32x4 g2 = { id[0], id[1], id[2], id[3] };  // row_index_0..3 (32-bit mode)
    i32x4 g3 = { id[4], id[5], id[6], id[7] };  // row_index_4..7
    i32x8 g4 = { 0, 0, 0, 0, 0, 0, 0, 0 };      // zero-filled (probe-verified form)
    __builtin_amdgcn_tensor_load_to_lds(g0, g1, g2, g3, g4, 0);
  };

  // Software pipeline: prefetch group g+1 via TDM while reducing group g.
  if (len > 0) tdm_issue(0, 0);
  int j = 0, bsel = 0;
  while (j < len) {
    const int n = min(ROWS_PER_TDM, len - j);
    const int jn = j + ROWS_PER_TDM;
    if (jn < len) {
      tdm_issue(jn, bsel ^ 1);
      __builtin_amdgcn_s_wait_tensorcnt(1);  // current buffer's DMA done
    } else {
      __builtin_amdgcn_s_wait_tensorcnt(0);  // drain tail
    }
    const float* rp = stage + (size_t)wave * 2 * ROWS_PER_TDM * SVDPP_D +
                      (size_t)bsel * ROWS_PER_TDM * SVDPP_D + lane * 4;
    for (int r = 0; r < n; ++r) {
      const float4 v = *reinterpret_cast<const float4*>(rp + r * SVDPP_D);
      ax += v.x; ay += v.y; az += v.z; aw += v.w;
    }
    j = jn; bsel ^= 1;
  }
#else
  // Fallback: direct vectorized gather (global_load_b128), L2-resident table.
  for (int jj = 0; jj < len; ++jj) {
    const int idx = hrow[jj];
    const float4 v = *reinterpret_cast<const float4*>(
        implicit_emb + (size_t)idx * SVDPP_D + lane * 4);
    ax += v.x; ay += v.y; az += v.z; aw += v.w;
  }
#endif

  const float norm = (len > 0) ? rsqrtf((float)len) : 0.f;
  const float ex = p.x + norm * ax, ey = p.y + norm * ay;
  const float ez = p.z + norm * az, ew = p.w + norm * aw;
  float partial = ex * q.x + ey * q.y + ez * q.z + ew * q.w;
#pragma unroll
  for (int off = 16; off > 0; off >>= 1)
    partial += __shfl_xor(partial, off, 32);

  if (lane == 0)
    out[b] = global_bias[0] + user_bias[uid] + item_bias[iid] + partial;
}

extern "C" void kernel_launch(void* const* d_in, const int* in_sizes, int n_in,
                              void* d_out, int out_size, void* d_ws, size_t ws_size,
                              hipStream_t stream) {
  const int*   user_ids     = (const int*)d_in[0];
  const int*   item_ids     = (const int*)d_in[1];
  const int*   hist_items   = (const int*)d_in[2];
  const int*   hist_len     = (const int*)d_in[3];
  const float* user_emb     = (const float*)d_in[4];
  const float* item_emb     = (const float*)d_in[5];
  const float* implicit_emb = (const float*)d_in[6];
  const float* user_bias    = (const float*)d_in[7];
  const float* item_bias    = (const float*)d_in[8];
  const float* global_bias  = (const float*)d_in[9];
  float* out = (float*)d_out;

  const int batch = in_sizes[0];
  const int n_items = in_sizes[6] / SVDPP_D;

  dim3 grid((batch + WAVES_PER_BLOCK - 1) / WAVES_PER_BLOCK);
  svdpp_score_kernel<<<grid, BLOCK_THREADS, 0, stream>>>(
      user_ids, item_ids, hist_items, hist_len,
      user_emb, item_emb, implicit_emb,
      user_bias, item_bias, global_bias,
      out, batch, n_items);
}